// NTMMemory_47708496724184
// MI455X (gfx1250) — compile-verified
//
#include <hip/hip_runtime.h>
#include <math.h>

#define B_ 32
#define N_ 16384
#define M_ 128
#define TR 8   // rows per async-staged tile in kernel 3

typedef __attribute__((ext_vector_type(2))) float v2f;
typedef __attribute__((ext_vector_type(8))) float v8f;

// ---------------- Kernel 0: zero r accumulator + reduction scratch ----------
__global__ __launch_bounds__(256) void k0_init(float* __restrict__ r_out,
                                               float* __restrict__ sums,
                                               int total_r, int total_s) {
  int i = blockIdx.x * blockDim.x + threadIdx.x;
  if (i < total_r) r_out[i] = 0.f;
  if (i < total_s) sums[i]  = 0.f;
}

// ---------------- Kernel 1: cosine similarity -> exp(beta*cos) --------------
// Grid: (B, N/128). Block: 256 threads = 8 waves, each wave owns 16 rows.
// Dot products via V_WMMA_F32_16X16X4_F32 accumulating over K=128 in steps of 4.
__global__ __launch_bounds__(256) void k1_similarity(
    const float* __restrict__ memory, const float* __restrict__ kvec,
    const float* __restrict__ beta, float* __restrict__ S_e,
    float* __restrict__ sumExp) {
  const int b    = blockIdx.x;
  const int base = blockIdx.y * 128;
  const int t    = threadIdx.x;
  const int lane = t & 31;
  const int wave = t >> 5;

  __shared__ float ksh[M_];
  __shared__ float s_nk;
  __shared__ float s_esum;

  if (t < M_) ksh[t] = kvec[b * M_ + t] + 1e-16f;
  if (t == 0) s_esum = 0.f;
  __syncthreads();
  if (wave == 0) {
    float p = 0.f;
    for (int i = lane; i < M_; i += 32) p += ksh[i] * ksh[i];
    for (int off = 16; off > 0; off >>= 1) p += __shfl_xor(p, off, 32);
    if (lane == 0) s_nk = fmaxf(sqrtf(p), 1e-8f);
  }
  __syncthreads();

  // A-matrix (16x4 f32) layout: lanes 0-15 rows 0-15 with K=0,1; lanes 16-31
  // same rows with K=2,3.  B (4x16) = broadcast of k chunk: lane j<16 holds
  // B[0..1][j]=k[kb..kb+1]; lanes 16-31 hold B[2..3][j]=k[kb+2..kb+3].
  const int hi      = lane >> 4;   // 0 or 1 (selects K=0,1 vs K=2,3)
  const int rloc    = lane & 15;
  const int rowTile = base + wave * 16;
  const float* Mb   = memory + ((size_t)b * N_ + rowTile + rloc) * (size_t)M_;

  v8f c = {};
  float nsq = 0.f;
#pragma unroll 4
  for (int ks = 0; ks < 32; ++ks) {
    const int kb = ks * 4 + hi * 2;
    v2f a, bb;
    a.x  = Mb[kb]     + 1e-16f;
    a.y  = Mb[kb + 1] + 1e-16f;
    bb.x = ksh[kb];
    bb.y = ksh[kb + 1];
    c = __builtin_amdgcn_wmma_f32_16x16x4_f32(false, a, false, bb,
                                              (short)0, c, false, false);
    nsq += a.x * a.x + a.y * a.y;
  }
  // full squared norm for row (lane&15): combine K halves across half-waves
  nsq += __shfl_xor(nsq, 16, 32);

  // D columns are all identical (== row dot).  Lane l<16 holds rows 0-7 in
  // c[0..7]; lanes 16-31 hold rows 8-15.  Lane l extracts row rr:
  const int sel = lane & 7;
  const int rr  = sel + (hi << 3);
  float d = c[0];
  if (sel == 1) d = c[1];
  if (sel == 2) d = c[2];
  if (sel == 3) d = c[3];
  if (sel == 4) d = c[4];
  if (sel == 5) d = c[5];
  if (sel == 6) d = c[6];
  if (sel == 7) d = c[7];

  const float nsq_r  = __shfl(nsq, rr, 32);
  const float bet    = beta[b];
  const bool  active = (lane & 8) == 0;  // lanes 0-7 -> rows 0-7, 16-23 -> 8-15

  float ebc = 0.f;
  if (active) {
    const float cosv = d / (fmaxf(sqrtf(nsq_r), 1e-8f) * s_nk);
    ebc = __expf(bet * cosv);             // beta<=10, cos<=1 -> no overflow
    S_e[(size_t)b * N_ + rowTile + rr] = ebc;
  }
  float wsum = ebc;                       // inactive lanes contribute 0
  for (int off = 16; off > 0; off >>= 1) wsum += __shfl_xor(wsum, off, 32);
  if (lane == 0) atomicAdd(&s_esum, wsum);
  __syncthreads();
  if (t == 0) atomicAdd(&sumExp[b], s_esum);
}

// ---------------- Kernel 2: interpolate + circular shift + sharpen ----------
// wg is elementwise, so each thread evaluates its 3 circular taps inline.
__global__ __launch_bounds__(256) void k2_shift_sharpen(
    const float* __restrict__ S_e, const float* __restrict__ w_prev,
    const float* __restrict__ g, const float* __restrict__ s,
    const float* __restrict__ gamma, const float* __restrict__ sumExp,
    float* __restrict__ S_wp, float* __restrict__ sumWp) {
  const int b  = blockIdx.x;
  const int n0 = blockIdx.y * 2048;
  const float invSum = 1.f / sumExp[b];
  const float gg  = g[b];
  const float omg = 1.f - gg;
  const float s0 = s[b * 3 + 0], s1 = s[b * 3 + 1], s2 = s[b * 3 + 2];
  const float gam = gamma[b];
  const float* Eb = S_e    + (size_t)b * N_;
  const float* Wb = w_prev + (size_t)b * N_;

  __shared__ float s_sum;
  if (threadIdx.x == 0) s_sum = 0.f;
  __syncthreads();

  float lsum = 0.f;
  for (int n = n0 + threadIdx.x; n < n0 + 2048; n += 256) {
    if (n + 256 < N_) {
      __builtin_prefetch(&Eb[n + 256], 0, 0);   // global_prefetch_b8
      __builtin_prefetch(&Wb[n + 256], 0, 0);
    }
    const int nm1 = (n == 0)      ? N_ - 1 : n - 1;
    const int np1 = (n == N_ - 1) ? 0      : n + 1;
    const float wgm = gg * Eb[nm1] * invSum + omg * Wb[nm1];
    const float wg0 = gg * Eb[n]   * invSum + omg * Wb[n];
    const float wgp = gg * Eb[np1] * invSum + omg * Wb[np1];
    const float wh  = s0 * wgm + s1 * wg0 + s2 * wgp;   // >= 0
    const float wp  = __powf(wh, gam);
    S_wp[(size_t)b * N_ + n] = wp;
    lsum += wp;
  }
  atomicAdd(&s_sum, lsum);
  __syncthreads();
  if (threadIdx.x == 0) atomicAdd(&sumWp[b], s_sum);
}

// ---------------- Kernel 3: normalize w, read r, erase/add write ------------
// Grid: (B, N/256). 256-row block chunk processed as 32 tiles of 8 rows.
// Each tile (4KB) is staged into LDS with GLOBAL_LOAD_ASYNC_TO_LDS_B128,
// double-buffered under ASYNCcnt: fetch of tile i+1 overlaps compute of tile i.
__global__ __launch_bounds__(256) void k3_read_write(
    const float* __restrict__ memory, const float* __restrict__ S_wp,
    const float* __restrict__ sumWp, const float* __restrict__ evec,
    const float* __restrict__ avec, float* __restrict__ r_out,
    float* __restrict__ w_out, float* __restrict__ newmem) {
  const int b     = blockIdx.x;
  const int n0    = blockIdx.y * 256;
  const int t     = threadIdx.x;
  const int m     = t & 127;
  const int rhalf = t >> 7;   // 0 or 1

  __shared__ float esh[M_], ash[M_];
  __shared__ float tile[2][TR * M_];   // 2 x 4KB double buffer
  __shared__ float red[256];

  if (t < M_) { esh[t] = evec[b * M_ + t]; ash[t] = avec[b * M_ + t]; }

  const float winv = 1.f / (sumWp[b] + 1e-16f);
  const float* Mb  = memory + ((size_t)b * N_ + n0) * (size_t)M_;
  float*       Ob  = newmem + ((size_t)b * N_ + n0) * (size_t)M_;
  const float* Pb  = S_wp + (size_t)b * N_ + n0;

  // 32-bit LDS byte offsets (flat LDS address: aperture in [63:32], offset low)
  const unsigned lds0 = (unsigned)(uintptr_t)(void*)&tile[0][0] + (unsigned)(t * 16);
  const unsigned lds1 = (unsigned)(uintptr_t)(void*)&tile[1][0] + (unsigned)(t * 16);

  // one async b128 per thread per tile: 256 lanes x 16B = 4KB tile
  auto issue_tile = [&](int ti, unsigned ldsoff) {
    const unsigned long long ga =
        (unsigned long long)(uintptr_t)(Mb + (size_t)ti * TR * M_ + t * 4);
    asm volatile("global_load_async_to_lds_b128 %0, %1, off"
                 :: "v"(ldsoff), "v"(ga) : "memory");
  };

  issue_tile(0, lds0);

  float racc = 0.f;
  const int NT = 256 / TR;   // 32 tiles
  for (int ti = 0; ti < NT; ++ti) {
    if (ti + 1 < NT) {
      issue_tile(ti + 1, (ti & 1) ? lds0 : lds1);
      asm volatile("s_wait_asynccnt 0x1" ::: "memory");  // tile ti complete
    } else {
      asm volatile("s_wait_asynccnt 0x0" ::: "memory");
    }
    __syncthreads();   // make all waves' async LDS writes visible

    const float* tb = &tile[ti & 1][0];
    const float  em = esh[m], am = ash[m];
#pragma unroll
    for (int r2 = 0; r2 < TR; r2 += 2) {
      const int rl = r2 + rhalf;
      const int n  = ti * TR + rl;
      const float wn = Pb[n] * winv;
      const float v  = tb[rl * M_ + m];
      Ob[(size_t)n * M_ + m] = v * (1.f - wn * em) + wn * am;
      racc += wn * v;
      if (m == 0) w_out[(size_t)b * N_ + n0 + n] = wn;
    }
    __syncthreads();   // all reads of this buffer done before it is refilled
  }

  red[t] = racc;
  __syncthreads();
  if (t < 128) atomicAdd(&r_out[b * M_ + t], red[t] + red[t + 128]);
}

// ---------------- Host launcher --------------------------------------------
extern "C" void kernel_launch(void* const* d_in, const int* in_sizes, int n_in,
                              void* d_out, int out_size, void* d_ws,
                              size_t ws_size, hipStream_t stream) {
  (void)in_sizes; (void)n_in; (void)out_size; (void)ws_size;
  const float* memory = (const float*)d_in[0];
  const float* k      = (const float*)d_in[1];
  const float* beta   = (const float*)d_in[2];
  const float* g      = (const float*)d_in[3];
  const float* s      = (const float*)d_in[4];
  const float* gamma  = (const float*)d_in[5];
  const float* w_prev = (const float*)d_in[6];
  const float* e      = (const float*)d_in[7];
  const float* a      = (const float*)d_in[8];

  float* out    = (float*)d_out;
  float* r_out  = out;                                   // [B, M]
  float* w_out  = out + (size_t)B_ * M_;                 // [B, N]
  float* nm_out = w_out + (size_t)B_ * N_;               // [B, N, M]

  float* ws     = (float*)d_ws;
  float* S_e    = ws;                                    // [B*N]
  float* S_wp   = ws + (size_t)B_ * N_;                  // [B*N]
  float* sumExp = ws + 2 * (size_t)B_ * N_;              // [B]
  // sumWp = sumExp + B_ (zeroed together)

  k0_init<<<dim3(16), dim3(256), 0, stream>>>(r_out, sumExp, B_ * M_, 2 * B_);
  k1_similarity<<<dim3(B_, N_ / 128), dim3(256), 0, stream>>>(
      memory, k, beta, S_e, sumExp);
  k2_shift_sharpen<<<dim3(B_, N_ / 2048), dim3(256), 0, stream>>>(
      S_e, w_prev, g, s, gamma, sumExp, S_wp, sumExp + B_);
  k3_read_write<<<dim3(B_, N_ / 256), dim3(256), 0, stream>>>(
      memory, S_wp, sumExp + B_, e, a, r_out, w_out, nm_out);
}